// ProbabilisticSurfaceDistance_16166256902864
// MI455X (gfx1250) — compile-verified
//
#include <hip/hip_runtime.h>
#include <hip/hip_bf16.h>

typedef __attribute__((ext_vector_type(2))) float v2f;
typedef __attribute__((ext_vector_type(8))) float v8f;
typedef __attribute__((ext_vector_type(4))) unsigned u32x4;
typedef __attribute__((ext_vector_type(8))) int     i32x8;
typedef __attribute__((ext_vector_type(4))) int     i32x4;

#define BLK 256
#define MAX_FS 4096
#define MAX_COLS 4096
#define TOPK 5

// ds_swizzle_b32 group-of-32 XOR pattern: offset = (xor<<10) | (or<<5) | and
#define SWZ_XOR(mask) (((mask) << 10) | (0x00 << 5) | 0x1f)

__device__ __forceinline__ float swz_min16(float v) {
    // min over the 16 lanes of each half-wave (xor masks < 16 stay in-half)
    float t;
    t = __int_as_float(__builtin_amdgcn_ds_swizzle(__float_as_int(v), SWZ_XOR(1)));
    v = fminf(v, t);
    t = __int_as_float(__builtin_amdgcn_ds_swizzle(__float_as_int(v), SWZ_XOR(2)));
    v = fminf(v, t);
    t = __int_as_float(__builtin_amdgcn_ds_swizzle(__float_as_int(v), SWZ_XOR(4)));
    v = fminf(v, t);
    t = __int_as_float(__builtin_amdgcn_ds_swizzle(__float_as_int(v), SWZ_XOR(8)));
    v = fminf(v, t);
    return v;
}

// ---- Tensor Data Mover bulk fill: copy ndwords*4 bytes from global to LDS. ----
// 1-D copy expressed as a D# descriptor (groups 0/1; groups 2/3 zero => <=2D).
// Call from exactly one wave; follows with s_wait_tensorcnt 0.
__device__ __forceinline__ void tdm_fill_lds(void* ldsDst, const void* gsrc, unsigned ndwords) {
#if __has_builtin(__builtin_amdgcn_tensor_load_to_lds) && __has_builtin(__builtin_amdgcn_s_wait_tensorcnt)
    unsigned lds_off = (unsigned)(size_t)ldsDst;      // flat shared addr low 32b == LDS offset
    unsigned long long ga = (unsigned long long)gsrc;
    u32x4 g0;
    g0[0] = 1u;                                        // count=1 (valid user descriptor)
    g0[1] = lds_off;                                   // lds_addr (bytes)
    g0[2] = (unsigned)(ga & 0xFFFFFFFFu);              // global_addr[31:0]
    g0[3] = (unsigned)((ga >> 32) & 0x01FFFFFFu)       // global_addr[56:32]
          | (2u << 30);                                // type=2 ("image")
    i32x8 g1;
    g1[0] = (int)(2u << 16);                           // data_size=2 -> 4 bytes
    g1[1] = (int)((ndwords & 0xFFFFu) << 16);          // tensor_dim0[15:0]  (bits 63:48)
    g1[2] = (int)((ndwords >> 16) | (1u << 16));       // tensor_dim0[31:16], tensor_dim1=1
    g1[3] = (int)((ndwords & 0xFFFFu) << 16);          // tile_dim0 (bits 127:112)
    g1[4] = 1;                                         // tile_dim1=1
    g1[5] = (int)ndwords;                              // tensor_dim0_stride[31:0]
    g1[6] = 0;
    g1[7] = 0;
    i32x4 z4 = {0, 0, 0, 0};
    i32x8 z8 = {0, 0, 0, 0, 0, 0, 0, 0};
    __builtin_amdgcn_tensor_load_to_lds(g0, g1, z4, z4, z8, 0);
    __builtin_amdgcn_s_wait_tensorcnt(0);
#else
    // fallback: plain per-lane loads (wave 0 only, 32 lanes)
    unsigned lane = (unsigned)(threadIdx.x & 31);
    float* d = (float*)ldsDst;
    const float* s = (const float*)gsrc;
    for (unsigned k = lane; k < ndwords; k += 32) d[k] = s[k];
#endif
}

__global__ void psd_zero_kernel(float* out) {
    if (threadIdx.x == 0 && blockIdx.x == 0) out[0] = 0.0f;
}

// Barycenters: SoA (for WMMA A-tiles) + packed column tiles + packed src for top-k.
__global__ __launch_bounds__(BLK)
void psd_bary_kernel(const float* __restrict__ sv, const int* __restrict__ sfaces,
                     const float* __restrict__ tv, const int* __restrict__ tfaces,
                     const float* __restrict__ fprobs, int Fs, int Ft,
                     float* __restrict__ sbx, float* __restrict__ sby,
                     float* __restrict__ sbz, float* __restrict__ sbn2,
                     float4* __restrict__ tpack, float4* __restrict__ spack) {
    int i = blockIdx.x * blockDim.x + threadIdx.x;
    const float third = 1.0f / 3.0f;
    if (i < Fs) {
        int a = sfaces[3*i], b = sfaces[3*i+1], c = sfaces[3*i+2];
        float x = (sv[3*a+0] + sv[3*b+0] + sv[3*c+0]) * third;
        float y = (sv[3*a+1] + sv[3*b+1] + sv[3*c+1]) * third;
        float z = (sv[3*a+2] + sv[3*b+2] + sv[3*c+2]) * third;
        sbx[i] = x; sby[i] = y; sbz[i] = z; sbn2[i] = x*x + y*y + z*z;
        spack[i] = make_float4(x, y, z, fprobs[i]);
    }
    if (i < Ft) {
        int a = tfaces[0*Ft+i], b = tfaces[1*Ft+i], c = tfaces[2*Ft+i];
        float x = (tv[3*a+0] + tv[3*b+0] + tv[3*c+0]) * third;
        float y = (tv[3*a+1] + tv[3*b+1] + tv[3*c+1]) * third;
        float z = (tv[3*a+2] + tv[3*b+2] + tv[3*c+2]) * third;
        // packed B-column: [-2x, -2y, -2z, |y|^2]
        tpack[i] = make_float4(-2.0f*x, -2.0f*y, -2.0f*z, x*x + y*y + z*z);
    }
}

// Sample NUM_PTS points per source face.
__global__ __launch_bounds__(BLK)
void psd_pts_kernel(const float* __restrict__ sv, const int* __restrict__ sfaces,
                    const float* __restrict__ fprobs,
                    const float* __restrict__ r1u, const float* __restrict__ r2u,
                    int Fs, int NP,
                    float* __restrict__ px, float* __restrict__ py,
                    float* __restrict__ pz, float* __restrict__ pn2,
                    float* __restrict__ pprob) {
    int tid = blockIdx.x * blockDim.x + threadIdx.x;
    if (tid >= Fs * NP) return;
    int f = tid / NP;
    float r1 = sqrtf(r1u[tid]);
    float r2 = r2u[tid];
    float w1 = 1.0f - r1, w2 = r1 * (1.0f - r2), w3 = r1 * r2;
    int a = sfaces[3*f], b = sfaces[3*f+1], c = sfaces[3*f+2];
    float x = w1*sv[3*a+0] + w2*sv[3*b+0] + w3*sv[3*c+0];
    float y = w1*sv[3*a+1] + w2*sv[3*b+1] + w3*sv[3*c+1];
    float z = w1*sv[3*a+2] + w2*sv[3*b+2] + w3*sv[3*c+2];
    px[tid] = x; py[tid] = y; pz[tid] = z; pn2[tid] = x*x + y*y + z*z;
    pprob[tid] = fprobs[f];
}

// WMMA core: out += sum_m w[m] * max(min_n (|r_m|^2 + |c_n|^2 - 2 r_m.c_n), 0).
// (clamp hoisted out of the loop: max(.,0) is monotone, commutes with min.)
// Columns staged in LDS once per block via TDM; each wave owns 2 row tiles
// (32 rows) and issues 2 WMMAs per shared 16-column tile (x2 unroll).
__global__ __launch_bounds__(BLK)
void psd_min_dist_wsum_kernel(const float* __restrict__ rx, const float* __restrict__ ry,
                              const float* __restrict__ rz, const float* __restrict__ rn2,
                              const float4* __restrict__ cpack,
                              const float* __restrict__ w, int nRows, int nCols,
                              float* __restrict__ out) {
    __shared__ float4 lcols[MAX_COLS];
    if (threadIdx.x < 32) tdm_fill_lds(lcols, cpack, (unsigned)nCols * 4u);
    __syncthreads();

    int gtid = blockIdx.x * blockDim.x + threadIdx.x;
    int wave = gtid >> 5;
    int lane = gtid & 31;
    int row0 = wave * 32;                       // two 16-row tiles per wave
    if (row0 >= nRows) return;                  // wave-uniform: EXEC stays all-ones

    bool hi = lane >= 16;
    int mA = row0 + (lane & 15);
    int mB = row0 + 16 + (lane & 15);
    v2f a0, a1;
    a0.x = hi ? rz[mA]  : rx[mA];   a0.y = hi ? rn2[mA] : ry[mA];
    a1.x = hi ? rz[mB]  : rx[mB];   a1.y = hi ? rn2[mB] : ry[mB];

    v8f vmin0 = {3.4e38f,3.4e38f,3.4e38f,3.4e38f,3.4e38f,3.4e38f,3.4e38f,3.4e38f};
    v8f vmin1 = vmin0;

#pragma unroll 2
    for (int t0 = 0; t0 < nCols; t0 += 16) {
        float4 cp = lcols[t0 + (lane & 15)];    // ds_load_b128, conflict-free
        v2f b;
        b.x = hi ? cp.z : cp.x;                 // K=2 : K=0 (already -2*coord)
        b.y = hi ? 1.0f : cp.y;                 // K=3 : K=1
        float yn2 = cp.w;
        v8f acc0 = {yn2, yn2, yn2, yn2, yn2, yn2, yn2, yn2};
        v8f acc1 = acc0;
        acc0 = __builtin_amdgcn_wmma_f32_16x16x4_f32(false, a0, false, b,
                                                     (short)0, acc0, false, false);
        acc1 = __builtin_amdgcn_wmma_f32_16x16x4_f32(false, a1, false, b,
                                                     (short)0, acc1, false, false);
#pragma unroll
        for (int r = 0; r < 8; ++r) {
            vmin0[r] = fminf(vmin0[r], acc0[r]);   // no per-tile clamp: hoisted
            vmin1[r] = fminf(vmin1[r], acc1[r]);
        }
    }

    // Row-min across the 16 lanes of each half-wave (ds_swizzle, imm pattern),
    // then the hoisted clamp to >= 0.
#pragma unroll
    for (int r = 0; r < 8; ++r) {
        vmin0[r] = fmaxf(swz_min16(vmin0[r]), 0.0f);
        vmin1[r] = fmaxf(swz_min16(vmin1[r]), 0.0f);
    }

    if ((lane & 15) == 0) {
        float s = 0.0f;
        int baseA = row0 + (hi ? 8 : 0);        // D row = vgpr + 8*(lane/16)
        int baseB = row0 + 16 + (hi ? 8 : 0);
#pragma unroll
        for (int r = 0; r < 8; ++r) {
            s += w[baseA + r] * vmin0[r];
            s += w[baseB + r] * vmin1[r];
        }
        atomicAdd(out, s);
    }
}

// Per-point top-6 nearest source barycenters with self-exclusion; K=5 mean.
// Source pack {x,y,z,prob} staged in LDS via TDM.
__global__ __launch_bounds__(BLK)
void psd_topk_kernel(const float* __restrict__ px, const float* __restrict__ py,
                     const float* __restrict__ pz, const float* __restrict__ pprob,
                     const float4* __restrict__ spack,
                     int Fs, int NP, int nPts, float* __restrict__ out) {
    __shared__ float4 ls[MAX_FS];
    if (threadIdx.x < 32) tdm_fill_lds(ls, spack, (unsigned)Fs * 4u);
    __syncthreads();

    int tid = blockIdx.x * blockDim.x + threadIdx.x;
    if (tid >= nPts) return;

    float qx = px[tid], qy = py[tid], qz = pz[tid];
    int own = tid / NP;

    float d0=3.4e38f,d1=3.4e38f,d2=3.4e38f,d3=3.4e38f,d4=3.4e38f,d5=3.4e38f;
    int   i0=-1,i1=-1,i2=-1,i3=-1,i4=-1,i5=-1;

    for (int j = 0; j < Fs; ++j) {
        float4 cp = ls[j];
        float dx = qx - cp.x, dy = qy - cp.y, dz = qz - cp.z;
        float d = dx*dx + dy*dy + dz*dz;
        if (d < d5) {
            d5 = d; i5 = j;
            if (d5 < d4) { float t=d4; d4=d5; d5=t; int u=i4; i4=i5; i5=u; }
            if (d4 < d3) { float t=d3; d3=d4; d4=t; int u=i3; i3=i4; i4=u; }
            if (d3 < d2) { float t=d2; d2=d3; d3=t; int u=i2; i2=i3; i3=u; }
            if (d2 < d1) { float t=d1; d1=d2; d2=t; int u=i1; i1=i2; i2=u; }
            if (d1 < d0) { float t=d0; d0=d1; d1=t; int u=i0; i0=i1; i1=u; }
        }
    }

    int ep = (i0==own)?0 : (i1==own)?1 : (i2==own)?2 :
             (i3==own)?3 : (i4==own)?4 : (i5==own)?5 : 6;
    float dd[6] = {d0,d1,d2,d3,d4,d5};
    int   ii[6] = {i0,i1,i2,i3,i4,i5};
    float acc = 0.0f;
#pragma unroll
    for (int t = 0; t < TOPK; ++t) {
        int p = t + ((t >= ep) ? 1 : 0);
        acc += ls[ii[p]].w * dd[p];             // prob lives in pack.w (LDS gather)
    }
    float pp = pprob[tid];
    atomicAdd(out, (1.0f - pp) * acc * (1.0f / (float)TOPK));
}

extern "C" void kernel_launch(void* const* d_in, const int* in_sizes, int n_in,
                              void* d_out, int out_size, void* d_ws, size_t ws_size,
                              hipStream_t stream) {
    const float* sv     = (const float*)d_in[0];
    const int*   sfaces = (const int*)  d_in[1];
    const float* tv     = (const float*)d_in[2];
    const int*   tfaces = (const int*)  d_in[3];
    const float* fprobs = (const float*)d_in[4];
    const float* r1u    = (const float*)d_in[5];
    const float* r2u    = (const float*)d_in[6];
    float* out = (float*)d_out;

    const int Fs   = in_sizes[4];
    const int Ft   = in_sizes[3] / 3;
    const int NP   = in_sizes[5] / Fs;
    const int nPts = Fs * NP;

    float* ws    = (float*)d_ws;
    float* sbx   = ws;            float* sby  = sbx  + Fs;
    float* sbz   = sby  + Fs;     float* sbn2 = sbz  + Fs;
    float* px    = sbn2 + Fs;     float* py   = px   + nPts;
    float* pz    = py   + nPts;   float* pn2  = pz   + nPts;
    float* pprb  = pn2  + nPts;
    // 16B-aligned packed arrays at the end (d_ws is 256B-aligned; counts are /16)
    float4* tpack = (float4*)(pprb + nPts);
    float4* spack = tpack + Ft;
    (void)ws_size; (void)n_in; (void)out_size;

    psd_zero_kernel<<<1, 32, 0, stream>>>(out);

    int nBary = (Fs > Ft ? Fs : Ft);
    psd_bary_kernel<<<(nBary + BLK - 1) / BLK, BLK, 0, stream>>>(
        sv, sfaces, tv, tfaces, fprobs, Fs, Ft,
        sbx, sby, sbz, sbn2, tpack, spack);

    psd_pts_kernel<<<(nPts + BLK - 1) / BLK, BLK, 0, stream>>>(
        sv, sfaces, fprobs, r1u, r2u, Fs, NP, px, py, pz, pn2, pprb);

    // forward loss: src_bc rows vs tgt columns, weights = face_probs
    {
        int threads = (Fs / 32) * 32;           // one wave per 32 rows
        psd_min_dist_wsum_kernel<<<(threads + BLK - 1) / BLK, BLK, 0, stream>>>(
            sbx, sby, sbz, sbn2, tpack, fprobs, Fs, Ft, out);
    }
    // reverse part 1: pts rows vs tgt columns, weights = pt_probs
    {
        int threads = (nPts / 32) * 32;
        psd_min_dist_wsum_kernel<<<(threads + BLK - 1) / BLK, BLK, 0, stream>>>(
            px, py, pz, pn2, tpack, pprb, nPts, Ft, out);
    }
    // reverse part 2: top-(K+1) over src_bc with self-exclusion
    psd_topk_kernel<<<(nPts + BLK - 1) / BLK, BLK, 0, stream>>>(
        px, py, pz, pprb, spack, Fs, NP, nPts, out);
}